// BatchDepthwiseConv_41085657153624
// MI455X (gfx1250) — compile-verified
//
#include <hip/hip_runtime.h>

// ---------------------------------------------------------------------------
// Batched depthwise 3x3 conv, per-sample filters.
// x:[32,128,128,128] f32  f:[32,3,3,128,1] f32 -> out:[32,126,126,128] f32
// Memory-bound (~0.53 GB min traffic @ 23.3 TB/s ~ 23us). No channel
// contraction -> WMMA not applicable; use CDNA5 async global->LDS DMA to
// exploit 9x stencil reuse from the 320KB WGP LDS.
// ---------------------------------------------------------------------------

#define BB  32
#define HH  128
#define WW  128
#define CC  128
#define HO  126
#define WO  126
#define TW  32            // output tile width
#define TIN (TW + 2)      // input tile width = 34
#define NTW ((WO + TW - 1) / TW)   // 4 tiles per output row

// gfx1250 async global->LDS copies (ASYNCcnt) --------------------------------
#if defined(__has_builtin)
#if __has_builtin(__builtin_amdgcn_global_load_async_to_lds_b128)
#define HAVE_ASYNC_LDS 1
#endif
#if __has_builtin(__builtin_amdgcn_s_wait_asynccnt)
#define HAVE_WAIT_ASYNC 1
#endif
#endif

typedef int v4i __attribute__((ext_vector_type(4)));
typedef __attribute__((address_space(1))) v4i v4i_glob;   // global (AS1)
typedef __attribute__((address_space(3))) v4i v4i_lds;    // LDS (AS3)

__device__ __forceinline__ void async_wait_all() {
#if defined(HAVE_WAIT_ASYNC)
  __builtin_amdgcn_s_wait_asynccnt(0);
#else
  asm volatile("s_wait_asynccnt 0x0" ::: "memory");
#endif
}

__global__ __launch_bounds__(256)
void BatchDepthwiseConv_kernel(const float* __restrict__ x,
                               const float* __restrict__ f,
                               float* __restrict__ out) {
  // 3 rows x 34 cols x 128 ch = 52224 bytes of LDS
  __shared__ __align__(16) float tile[3 * TIN * CC];

  const int blk    = blockIdx.x;
  const int tw_idx = blk % NTW;
  const int ho     = (blk / NTW) % HO;
  const int b      = blk / (NTW * HO);

  const int wo0     = tw_idx * TW;
  const int tw      = (WO - wo0 < TW) ? (WO - wo0) : TW;   // 32 or 30
  const int in_cols = tw + 2;                              // 34 or 32

  const int tid = threadIdx.x;

  // ---- Stage 3 input rows (contiguous w*c spans) into LDS, 16B per op ----
  const int chunks_per_row = in_cols * (CC / 4);           // float4 chunks
  const int total_chunks   = 3 * chunks_per_row;
  const size_t x_base = (((size_t)b * HH + (size_t)ho) * WW + (size_t)wo0) * CC;

  for (int i = tid; i < total_chunks; i += 256) {
    const int r = i / chunks_per_row;
    const int j = i - r * chunks_per_row;                  // float4 idx in row
    const float* src = x + x_base + (size_t)r * (WW * CC) + (size_t)j * 4;
    float*       dst = &tile[r * (TIN * CC) + j * 4];
#if defined(HAVE_ASYNC_LDS)
    __builtin_amdgcn_global_load_async_to_lds_b128(
        (v4i_glob*)src, (v4i_lds*)dst, /*offset=*/0, /*cpol=*/0);
#else
    *(float4*)dst = *(const float4*)src;
#endif
  }
#if defined(HAVE_ASYNC_LDS)
  async_wait_all();
#endif
  __syncthreads();

  // ---- Per-lane filter taps: fixed channel quad, 9 taps in registers ----
  const int lane_c = (tid & 31) * 4;                       // channel quad base
  const float* fb = f + ((size_t)b * 9) * CC + lane_c;     // f[b][kh][kw][c]
  float4 fr[9];
#pragma unroll
  for (int k = 0; k < 9; ++k)
    fr[k] = *(const float4*)(fb + (size_t)k * CC);

  // ---- Compute: 8 pixels in flight (tid/32), loop over tile width ----
  const int psub = tid >> 5;                               // 0..7
  for (int p = psub; p < tw; p += 8) {
    float4 acc = make_float4(0.f, 0.f, 0.f, 0.f);
#pragma unroll
    for (int kh = 0; kh < 3; ++kh) {
#pragma unroll
      for (int kw = 0; kw < 3; ++kw) {
        const float4 v =
            *(const float4*)&tile[(kh * TIN + p + kw) * CC + lane_c];
        const float4 w = fr[kh * 3 + kw];
        acc.x = fmaf(v.x, w.x, acc.x);
        acc.y = fmaf(v.y, w.y, acc.y);
        acc.z = fmaf(v.z, w.z, acc.z);
        acc.w = fmaf(v.w, w.w, acc.w);
      }
    }
    float* o = out + (((size_t)b * HO + ho) * WO + (wo0 + p)) * CC + lane_c;
    *(float4*)o = acc;
  }
}

extern "C" void kernel_launch(void* const* d_in, const int* in_sizes, int n_in,
                              void* d_out, int out_size, void* d_ws, size_t ws_size,
                              hipStream_t stream) {
  (void)in_sizes; (void)n_in; (void)d_ws; (void)ws_size; (void)out_size;
  const float* x = (const float*)d_in[0];
  const float* f = (const float*)d_in[1];
  float* out = (float*)d_out;

  const int grid = BB * HO * NTW;   // 32 * 126 * 4 = 16128 blocks
  BatchDepthwiseConv_kernel<<<grid, 256, 0, stream>>>(x, f, out);
}